// MambaIntegerBlock_13185549598991
// MI455X (gfx1250) — compile-verified
//
#include <hip/hip_runtime.h>
#include <hip/hip_bf16.h>

// ---------------------------------------------------------------------------
// MambaIntegerBlock on gfx1250 (MI455X), fp32 end-to-end.
// GEMMs via V_WMMA_F32_16X16X4_F32 (fp32 WMMA, matches reference numerics),
// software-pipelined (double-buffered fragments) with 2 M-tiles x 4 N-tiles
// per wave. Linear scan over L done as 3-pass segmented recurrence (exact).
// ---------------------------------------------------------------------------

typedef float v2f __attribute__((ext_vector_type(2)));
typedef float v8f __attribute__((ext_vector_type(8)));

#define Bc   2
#define Lc   1024
#define DMc  768
#define DIc  1536
#define NSc  16
#define DTRc 48
#define XDBL 80          // DTR + 2*NS
#define ROWS (Bc*Lc)     // 2048
#define SEGS 16
#define SEGLEN (Lc/SEGS) // 64

__device__ __forceinline__ float dsp_f(float v) {
    return 0.25f * (v + sqrtf(v * v + 4.0f));
}

// ---------------- norm: hs = (x - mean) * 2^-k * gamma ----------------------
__global__ __launch_bounds__(256) void norm_kernel(const float* __restrict__ x,
                                                   const float* __restrict__ gamma,
                                                   float* __restrict__ hs) {
    __shared__ float red[256];
    const int row = blockIdx.x;                  // 0..2047
    const float* xr = x + (size_t)row * DMc;
    const int t = threadIdx.x;
    float v0 = xr[t], v1 = xr[t + 256], v2 = xr[t + 512];
    red[t] = v0 + v1 + v2;
    __syncthreads();
    for (int off = 128; off > 0; off >>= 1) {
        if (t < off) red[t] += red[t + off];
        __syncthreads();
    }
    const float mean = red[0] * (1.0f / 768.0f);
    __syncthreads();
    float d0 = v0 - mean, d1 = v1 - mean, d2 = v2 - mean;
    red[t] = d0 * d0 + d1 * d1 + d2 * d2;
    __syncthreads();
    for (int off = 128; off > 0; off >>= 1) {
        if (t < off) red[t] += red[t + off];
        __syncthreads();
    }
    const float var = red[0] * (1.0f / 768.0f);
    // k = max(round(log2(sqrt(var+1e-9))), 0); scale = 2^-k
    const float k = fmaxf(rintf(0.5f * log2f(var + 1e-9f)), 0.0f);
    const float scale = exp2f(-k);
    float* o = hs + (size_t)row * DMc;
    o[t]       = d0 * scale * gamma[t];
    o[t + 256] = d1 * scale * gamma[t + 256];
    o[t + 512] = d2 * scale * gamma[t + 512];
}

// ---------------- fp32 WMMA GEMM:  C[m,n] = sum_k A[m,k]*W[n,k] + bias[n] ---
// A: MxK row-major (lda), W: NxK row-major (ldw). One wave computes a
// (16*MT) x (16*NT) strip, k-loop double-buffered so next-step fragment loads
// overlap the current WMMAs. EPI==1: C = residual + clip(C,+-32000)*gate.
template <int NT, int MT, int EPI>
__global__ __launch_bounds__(128) void wmma_gemm_kernel(
    const float* __restrict__ A, int lda,
    const float* __restrict__ W, int ldw,
    const float* __restrict__ bias,
    float* __restrict__ C, int ldc,
    int M, int N, int K,
    const float* __restrict__ residual,
    const float* __restrict__ gatep) {
    const int wave   = blockIdx.x * (blockDim.x >> 5) + (threadIdx.x >> 5);
    const int tilesM = M / (16 * MT);
    const int groups = N / (16 * NT);
    if (wave >= tilesM * groups) return;          // whole-wave uniform exit
    const int tm = wave % tilesM;
    const int tg = wave / tilesM;
    const int m0 = tm * (16 * MT);
    const int n0 = tg * (16 * NT);
    const int lane = threadIdx.x & 31;
    const int row  = lane & 15;                   // M row (A) / N col (B frag)
    const int kh   = lane >> 4;                   // K half: lanes16-31 -> K=2,3

    const float* Ap[MT];
    const float* Wp[NT];
#pragma unroll
    for (int i = 0; i < MT; ++i) Ap[i] = A + (size_t)(m0 + i * 16 + row) * lda + 2 * kh;
#pragma unroll
    for (int j = 0; j < NT; ++j) Wp[j] = W + (size_t)(n0 + j * 16 + row) * ldw + 2 * kh;

    v8f acc[MT][NT] = {};
    v2f aCur[MT], bCur[NT];
#pragma unroll
    for (int i = 0; i < MT; ++i) aCur[i] = *(const v2f*)(Ap[i]);
#pragma unroll
    for (int j = 0; j < NT; ++j) bCur[j] = *(const v2f*)(Wp[j]);

#pragma unroll 4
    for (int k = 0; k < K; k += 4) {
        const int kn = (k + 4 < K) ? (k + 4) : 0;   // wrap: harmless dummy load
        v2f aNxt[MT], bNxt[NT];
#pragma unroll
        for (int i = 0; i < MT; ++i) aNxt[i] = *(const v2f*)(Ap[i] + kn);
#pragma unroll
        for (int j = 0; j < NT; ++j) bNxt[j] = *(const v2f*)(Wp[j] + kn);
#pragma unroll
        for (int i = 0; i < MT; ++i)
#pragma unroll
            for (int j = 0; j < NT; ++j)
                acc[i][j] = __builtin_amdgcn_wmma_f32_16x16x4_f32(
                    false, aCur[i], false, bCur[j], (short)0, acc[i][j], false, false);
#pragma unroll
        for (int i = 0; i < MT; ++i) aCur[i] = aNxt[i];
#pragma unroll
        for (int j = 0; j < NT; ++j) bCur[j] = bNxt[j];
    }

    const float g = (EPI == 1) ? gatep[0] : 0.0f;
#pragma unroll
    for (int i = 0; i < MT; ++i) {
#pragma unroll
        for (int j = 0; j < NT; ++j) {
            const int n = n0 + j * 16 + row;
            const float bn = bias[n];
#pragma unroll
            for (int r = 0; r < 8; ++r) {
                const int m = m0 + i * 16 + kh * 8 + r;  // C layout: VGPR r, lane-half kh
                float val = acc[i][j][r] + bn;
                if (EPI == 1)
                    val = fminf(fmaxf(val, -32000.0f), 32000.0f) * g +
                          residual[(size_t)m * ldc + n];
                C[(size_t)m * ldc + n] = val;
            }
        }
    }
}

// ---------------- depthwise causal conv(k=4) + bias + dsp -------------------
__global__ __launch_bounds__(256) void conv_dsp_kernel(const float* __restrict__ xz,
                                                       const float* __restrict__ cw,
                                                       const float* __restrict__ cb,
                                                       float* __restrict__ xo) {
    const size_t gid = (size_t)blockIdx.x * 256 + threadIdx.x;  // B*L*DI
    const int c = (int)(gid % DIc);
    const int l = (int)((gid / DIc) % Lc);
    const int b = (int)(gid / ((size_t)DIc * Lc));
    const float w0 = cw[c * 4 + 0], w1 = cw[c * 4 + 1],
                w2 = cw[c * 4 + 2], w3 = cw[c * 4 + 3];
    const float* base = xz + (size_t)b * Lc * (2 * DIc) + c;    // x half of xz
    float acc = cb[c];
    if (l >= 3) acc += w0 * base[(size_t)(l - 3) * (2 * DIc)];
    if (l >= 2) acc += w1 * base[(size_t)(l - 2) * (2 * DIc)];
    if (l >= 1) acc += w2 * base[(size_t)(l - 1) * (2 * DIc)];
    acc += w3 * base[(size_t)l * (2 * DIc)];
    xo[gid] = dsp_f(acc);
}

// ---------------- scan pass 1: per-segment (prod a, local end) --------------
__global__ __launch_bounds__(256) void scan_pass1(
    const float* __restrict__ dmod, const float* __restrict__ xo,
    const float* __restrict__ xdbl,
    const float* __restrict__ bnum, const int* __restrict__ shifts,
    float* __restrict__ segP, float* __restrict__ segE) {
    const int gid = blockIdx.x * 256 + threadIdx.x;   // b*(S*DI)+seg*DI+d
    const int d   = gid % DIc;
    const int seg = (gid / DIc) % SEGS;
    const int b   = gid / (DIc * SEGS);
    float bn[NSc], sc[NSc], h[NSc], P[NSc];
#pragma unroll
    for (int n = 0; n < NSc; ++n) {
        bn[n] = bnum[d * NSc + n];
        sc[n] = exp2f(-(float)shifts[d * NSc + n]);
        h[n] = 0.0f; P[n] = 1.0f;
    }
    const int t0 = seg * SEGLEN;
    for (int tt = 0; tt < SEGLEN; ++tt) {
        const size_t rowi = (size_t)b * Lc + (t0 + tt);
        const float dm = dmod[rowi * DIc + d];
        const float xv = xo[rowi * DIc + d];
        const float xact = xv * (dsp_f(dm) * 0.1f);
        const float4* xb = (const float4*)(xdbl + rowi * XDBL + DTRc);
        const float4 B0 = xb[0], B1 = xb[1], B2 = xb[2], B3 = xb[3];
        const float Bv[NSc] = {B0.x, B0.y, B0.z, B0.w, B1.x, B1.y, B1.z, B1.w,
                               B2.x, B2.y, B2.z, B2.w, B3.x, B3.y, B3.z, B3.w};
#pragma unroll
        for (int n = 0; n < NSc; ++n) {
            const float a = fminf(fmaxf(bn[n] + dm, 0.0f), 32767.0f) * sc[n];
            h[n] = a * h[n] + xact * Bv[n];
            P[n] *= a;
        }
    }
    const size_t o = (size_t)gid * NSc;
#pragma unroll
    for (int n = 0; n < NSc; ++n) { segP[o + n] = P[n]; segE[o + n] = h[n]; }
}

// ---------------- scan pass 2: stitch segment-entry states ------------------
__global__ __launch_bounds__(256) void scan_pass2(const float* __restrict__ segP,
                                                  const float* __restrict__ segE,
                                                  float* __restrict__ h0v) {
    const int gid = blockIdx.x * 256 + threadIdx.x;   // B*DI*NS
    const int n = gid % NSc;
    const int d = (gid / NSc) % DIc;
    const int b = gid / (NSc * DIc);
    float h = 0.0f;
    for (int seg = 0; seg < SEGS; ++seg) {
        const size_t idx = ((((size_t)b * SEGS + seg) * DIc + d) * NSc) + n;
        h0v[idx] = h;
        h = segE[idx] + segP[idx] * h;
    }
}

// ---------------- scan pass 3: exact recurrence + y + gate ------------------
__global__ __launch_bounds__(256) void scan_pass3(
    const float* __restrict__ dmod, const float* __restrict__ xo,
    const float* __restrict__ xdbl, const float* __restrict__ xz,
    const float* __restrict__ bnum, const int* __restrict__ shifts,
    const float* __restrict__ h0v, float* __restrict__ yg) {
    const int gid = blockIdx.x * 256 + threadIdx.x;
    const int d   = gid % DIc;
    const int seg = (gid / DIc) % SEGS;
    const int b   = gid / (DIc * SEGS);
    float bn[NSc], sc[NSc], h[NSc];
    const size_t o = (size_t)gid * NSc;
#pragma unroll
    for (int n = 0; n < NSc; ++n) {
        bn[n] = bnum[d * NSc + n];
        sc[n] = exp2f(-(float)shifts[d * NSc + n]);
        h[n] = h0v[o + n];
    }
    const int t0 = seg * SEGLEN;
    for (int tt = 0; tt < SEGLEN; ++tt) {
        const size_t rowi = (size_t)b * Lc + (t0 + tt);
        const float dm = dmod[rowi * DIc + d];
        const float xv = xo[rowi * DIc + d];
        const float xact = xv * (dsp_f(dm) * 0.1f);
        const float z = xz[rowi * (2 * DIc) + DIc + d];
        const float4* xb = (const float4*)(xdbl + rowi * XDBL + DTRc);
        const float4 B0 = xb[0], B1 = xb[1], B2 = xb[2], B3 = xb[3];
        const float4 C0 = xb[4], C1 = xb[5], C2 = xb[6], C3 = xb[7];
        const float Bv[NSc] = {B0.x, B0.y, B0.z, B0.w, B1.x, B1.y, B1.z, B1.w,
                               B2.x, B2.y, B2.z, B2.w, B3.x, B3.y, B3.z, B3.w};
        const float Cv[NSc] = {C0.x, C0.y, C0.z, C0.w, C1.x, C1.y, C1.z, C1.w,
                               C2.x, C2.y, C2.z, C2.w, C3.x, C3.y, C3.z, C3.w};
        float y = 0.0f;
#pragma unroll
        for (int n = 0; n < NSc; ++n) {
            const float a = fminf(fmaxf(bn[n] + dm, 0.0f), 32767.0f) * sc[n];
            h[n] = a * h[n] + xact * Bv[n];
            const float hc = fminf(fmaxf(h[n], -32000.0f), 32000.0f);
            y += hc * Cv[n];
        }
        const float gate = 0.5f * (1.0f + z / sqrtf(z * z + 1.0f));
        yg[rowi * DIc + d] = y * gate;
    }
}

// ---------------------------------------------------------------------------
extern "C" void kernel_launch(void* const* d_in, const int* in_sizes, int n_in,
                              void* d_out, int out_size, void* d_ws, size_t ws_size,
                              hipStream_t stream) {
    const float* hidden = (const float*)d_in[0];
    const float* gamma  = (const float*)d_in[1];
    const float* W_in   = (const float*)d_in[2];
    const float* b_in   = (const float*)d_in[3];
    const float* conv_w = (const float*)d_in[4];
    const float* conv_b = (const float*)d_in[5];
    const float* W_x    = (const float*)d_in[6];
    const float* b_x    = (const float*)d_in[7];
    const float* W_dt   = (const float*)d_in[8];
    const float* b_dt   = (const float*)d_in[9];
    const float* W_out  = (const float*)d_in[10];
    const float* b_out  = (const float*)d_in[11];
    const float* bnum   = (const float*)d_in[12];
    const float* resg   = (const float*)d_in[13];
    const int*   shifts = (const int*)d_in[14];
    float* out = (float*)d_out;

    float* ws   = (float*)d_ws;
    float* hs   = ws;                                    // 2048*768
    float* xz   = hs   + (size_t)ROWS * DMc;             // 2048*3072
    float* xo   = xz   + (size_t)ROWS * 2 * DIc;         // 2048*1536
    float* xdbl = xo   + (size_t)ROWS * DIc;             // 2048*80
    float* dmod = xdbl + (size_t)ROWS * XDBL;            // 2048*1536
    float* yg   = dmod + (size_t)ROWS * DIc;             // 2048*1536
    float* segP = yg   + (size_t)ROWS * DIc;             // 2*16*1536*16
    float* segE = segP + (size_t)Bc * SEGS * DIc * NSc;
    float* h0v  = segE + (size_t)Bc * SEGS * DIc * NSc;

    norm_kernel<<<ROWS, 256, 0, stream>>>(hidden, gamma, hs);
    // xz = hs @ W_in^T + b_in (2048x768x3072): 64 M-tiles * 48 groups = 3072 waves
    wmma_gemm_kernel<4, 2, 0><<<768, 128, 0, stream>>>(hs, DMc, W_in, DMc, b_in,
                                                       xz, 2 * DIc, ROWS, 2 * DIc, DMc,
                                                       nullptr, nullptr);
    conv_dsp_kernel<<<(ROWS * DIc) / 256, 256, 0, stream>>>(xz, conv_w, conv_b, xo);
    // x_dbl = x @ W_x^T + b_x (2048x1536x80): 64*5 = 320 waves
    wmma_gemm_kernel<1, 2, 0><<<80, 128, 0, stream>>>(xo, DIc, W_x, DIc, b_x,
                                                      xdbl, XDBL, ROWS, XDBL, DIc,
                                                      nullptr, nullptr);
    // decay_mod = dt @ W_dt^T + b_dt (2048x48x1536): 64*24 = 1536 waves
    wmma_gemm_kernel<4, 2, 0><<<384, 128, 0, stream>>>(xdbl, XDBL, W_dt, DTRc, b_dt,
                                                       dmod, DIc, ROWS, DIc, DTRc,
                                                       nullptr, nullptr);
    scan_pass1<<<(Bc * SEGS * DIc) / 256, 256, 0, stream>>>(dmod, xo, xdbl, bnum,
                                                            shifts, segP, segE);
    scan_pass2<<<(Bc * DIc * NSc) / 256, 256, 0, stream>>>(segP, segE, h0v);
    scan_pass3<<<(Bc * SEGS * DIc) / 256, 256, 0, stream>>>(dmod, xo, xdbl, xz, bnum,
                                                            shifts, h0v, yg);
    // out = residual + clip(yg @ W_out^T + b_out)*res_gate (2048x1536x768)
    wmma_gemm_kernel<4, 2, 1><<<192, 128, 0, stream>>>(yg, DIc, W_out, DIc, b_out,
                                                       out, DMc, ROWS, DMc, DIc,
                                                       hidden, resg);
}